// ADE_nl_Trajectories_coarse_72499047957062
// MI455X (gfx1250) — compile-verified
//
#include <hip/hip_runtime.h>
#include <hip/hip_bf16.h>
#include <math.h>

// ---------------------------------------------------------------------------
// ADE over "non-linear" trajectory columns.
// Memory-bound: 288 MB streamed -> ~12.4 us floor at 23.3 TB/s on MI455X.
// One thread processes 2 adjacent columns (float4 = b128 coalesced loads).
// Block partials -> d_ws; single-block finalize -> d_out[0].
// Wave reductions use V_WMMA_F32_16X16X4_F32 (sum-by-ones) + shfl_xor(16).
// ---------------------------------------------------------------------------

typedef __attribute__((ext_vector_type(2))) float v2f;
typedef __attribute__((ext_vector_type(8))) float v8f;

#define T_STEPS 12
#define BLOCK   256
#define WAVES   (BLOCK / 32)

// Exact f32 sum of one value per lane across a wave32, via WMMA:
//   A (16x4): VGPR0 carries v[lane] -> rows 0..15 get {v[m], v[m+16]} in two
//   K slots, VGPR1 = 0. B = all-ones (layout irrelevant: every element is 1).
//   D[m][*] = v[m] + v[m+16]. Summing D's 8 VGPRs per lane gives
//   sum(v[0..7]+v[16..23]) in lanes 0..15 and sum(v[8..15]+v[24..31]) in
//   lanes 16..31; one xor-16 shuffle completes the wave total (all lanes).
__device__ __forceinline__ float wave_sum_wmma(float v) {
    v2f A = {v, 0.0f};
    v2f B = {1.0f, 1.0f};
    v8f c = {};
    v8f d = __builtin_amdgcn_wmma_f32_16x16x4_f32(
        /*neg_a=*/false, A, /*neg_b=*/false, B,
        /*c_mod=*/(short)0, c, /*reuse_a=*/false, /*reuse_b=*/false);
    float s = d[0] + d[1] + d[2] + d[3] + d[4] + d[5] + d[6] + d[7];
    s += __shfl_xor(s, 16, 32);
    return s;
}

// One Menger-curvature step for a single column: points (x0,y0),(x1,y1),(x2,y2).
__device__ __forceinline__ void curv_step(float x0, float y0, float x1, float y1,
                                          float x2, float y2,
                                          int& run, int& mx, int& msum) {
    float area2 = fabsf(x0 * (y1 - y2) + x1 * (y2 - y0) + x2 * (y0 - y1));
    float dx01 = x1 - x0, dy01 = y1 - y0;
    float dx12 = x2 - x1, dy12 = y2 - y1;
    float dx02 = x2 - x0, dy02 = y2 - y0;
    float dist = sqrtf(dx01 * dx01 + dy01 * dy01) *
                 sqrtf(dx12 * dx12 + dy12 * dy12) *
                 sqrtf(dx02 * dx02 + dy02 * dy02);
    float k = 2.0f * area2 / dist;                 // NaN only when 0/0
    int m = (!__builtin_isnan(k) && fabsf(k) >= 1.0f) ? 1 : 0;  // NaN->0, inf passes
    run = m ? run + 1 : 0;
    mx = run > mx ? run : mx;
    msum += m;
}

__global__ __launch_bounds__(BLOCK)
void traj_partial_kernel(const float* __restrict__ input,
                         const float* __restrict__ target,
                         float2* __restrict__ partials,
                         int N, int nPairsFull, int tail) {
    const int p = blockIdx.x * BLOCK + threadIdx.x;
    const long long rowStride = (long long)N * 2;   // floats per t-plane

    float a = 0.0f;          // sel*ade contribution of this thread
    bool sel0 = false, sel1 = false;

    if (p < nPairsFull) {
        // --- fast path: two adjacent columns via float4 (b128) loads ---
        const long long base = (long long)p * 4;
        float4 tm2, tm1;
        float ade0 = 0.0f, ade1 = 0.0f;
        int run0 = 0, mx0 = 0, sum0 = 0;
        int run1 = 0, mx1 = 0, sum1 = 0;
#pragma unroll
        for (int t = 0; t < T_STEPS; ++t) {
            const float4 tg = *(const float4*)(target + (long long)t * rowStride + base);
            const float4 in = *(const float4*)(input  + (long long)t * rowStride + base);
            float dx0 = in.x - tg.x, dy0 = in.y - tg.y;
            float dx1 = in.z - tg.z, dy1 = in.w - tg.w;
            ade0 += sqrtf(dx0 * dx0 + dy0 * dy0);
            ade1 += sqrtf(dx1 * dx1 + dy1 * dy1);
            if (t >= 2) {
                curv_step(tm2.x, tm2.y, tm1.x, tm1.y, tg.x, tg.y, run0, mx0, sum0);
                curv_step(tm2.z, tm2.w, tm1.z, tm1.w, tg.z, tg.w, run1, mx1, sum1);
            }
            tm2 = tm1;
            tm1 = tg;
        }
        sel0 = (mx0 >= 3) && (mx0 < 11) && (sum0 > 0);
        sel1 = (mx1 >= 3) && (mx1 < 11) && (sum1 > 0);
        const float inv_t = 1.0f / (float)T_STEPS;
        if (sel0) a += ade0 * inv_t;
        if (sel1) a += ade1 * inv_t;
    } else if (tail && p == nPairsFull) {
        // --- remainder column (only if N is odd) ---
        const long long base = (long long)(N - 1) * 2;
        float2 tm2, tm1;
        float ade0 = 0.0f;
        int run0 = 0, mx0 = 0, sum0 = 0;
#pragma unroll
        for (int t = 0; t < T_STEPS; ++t) {
            const float2 tg = *(const float2*)(target + (long long)t * rowStride + base);
            const float2 in = *(const float2*)(input  + (long long)t * rowStride + base);
            float dx0 = in.x - tg.x, dy0 = in.y - tg.y;
            ade0 += sqrtf(dx0 * dx0 + dy0 * dy0);
            if (t >= 2)
                curv_step(tm2.x, tm2.y, tm1.x, tm1.y, tg.x, tg.y, run0, mx0, sum0);
            tm2 = tm1;
            tm1 = tg;
        }
        sel0 = (mx0 >= 3) && (mx0 < 11) && (sum0 > 0);
        if (sel0) a += ade0 * (1.0f / (float)T_STEPS);
    }
    // (threads past the data contribute a = 0, sel = false)

    // ---- block reduction: WMMA wave sums + exact ballot counts ----
    const float aw = wave_sum_wmma(a);                       // full EXEC here
    const unsigned long long m0 = __ballot(sel0);
    const unsigned long long m1 = __ballot(sel1);
    const float cw = (float)(__popcll(m0) + __popcll(m1));

    __shared__ float sA[WAVES];
    __shared__ float sC[WAVES];
    const int wid = threadIdx.x >> 5;
    if ((threadIdx.x & 31) == 0) {
        sA[wid] = aw;
        sC[wid] = cw;
    }
    __syncthreads();
    if (threadIdx.x == 0) {
        float A = 0.0f, C = 0.0f;
#pragma unroll
        for (int i = 0; i < WAVES; ++i) { A += sA[i]; C += sC[i]; }
        partials[blockIdx.x] = make_float2(A, C);
    }
}

__global__ __launch_bounds__(BLOCK)
void traj_finalize_kernel(const float2* __restrict__ partials, int nPart,
                          float* __restrict__ out) {
    float a = 0.0f, c = 0.0f;
    for (int i = threadIdx.x; i < nPart; i += BLOCK) {
        const float2 v = partials[i];
        a += v.x;
        c += v.y;
    }
    const float aw = wave_sum_wmma(a);   // loop reconverged: full EXEC
    const float cw = wave_sum_wmma(c);

    __shared__ float sA[WAVES];
    __shared__ float sC[WAVES];
    const int wid = threadIdx.x >> 5;
    if ((threadIdx.x & 31) == 0) {
        sA[wid] = aw;
        sC[wid] = cw;
    }
    __syncthreads();
    if (threadIdx.x == 0) {
        float A = 0.0f, C = 0.0f;
#pragma unroll
        for (int i = 0; i < WAVES; ++i) { A += sA[i]; C += sC[i]; }
        out[0] = (C > 0.0f) ? (A / fmaxf(C, 1.0f)) : 0.0f;
    }
}

extern "C" void kernel_launch(void* const* d_in, const int* in_sizes, int n_in,
                              void* d_out, int out_size, void* d_ws, size_t ws_size,
                              hipStream_t stream) {
    const float* input  = (const float*)d_in[0];   // (T, N, 2) f32
    const float* target = (const float*)d_in[1];   // (T, N, 2) f32

    const long long total = (long long)in_sizes[0];
    const int N = (int)(total / (T_STEPS * 2));    // 1,500,000 columns
    const int nPairsFull = N >> 1;
    const int tail = N & 1;
    const int nThreads = nPairsFull + tail;
    const int nBlocks = (nThreads + BLOCK - 1) / BLOCK;

    float2* partials = (float2*)d_ws;              // nBlocks * 8 bytes (~23 KB)

    traj_partial_kernel<<<dim3(nBlocks), dim3(BLOCK), 0, stream>>>(
        input, target, partials, N, nPairsFull, tail);
    traj_finalize_kernel<<<dim3(1), dim3(BLOCK), 0, stream>>>(
        partials, nBlocks, (float*)d_out);
}